// custom_model_60292750901743
// MI455X (gfx1250) — compile-verified
//
#include <hip/hip_runtime.h>
#include <math.h>

#define DT_F    0.02f
#define N_STEPS 50

// gfx1250 hardware tanh (new TRANS op on CDNA5). Builtin preferred so the
// backend handles the TRANS data hazard; asm fallback inserts v_nop per the
// "1 independent op or V_NOP after trans op" rule.
#if defined(__gfx1250__) && __has_builtin(__builtin_amdgcn_tanhf)
#warning "tanh path: __builtin_amdgcn_tanhf (single v_tanh_f32)"
#elif defined(__AMDGCN__)
#warning "tanh path: inline-asm v_tanh_f32 + v_nop"
#endif
__device__ __forceinline__ float tanh_fast(float x) {
#if defined(__gfx1250__) && __has_builtin(__builtin_amdgcn_tanhf)
    return __builtin_amdgcn_tanhf(x);
#elif defined(__AMDGCN__)
    float r;
    asm volatile("v_tanh_f32 %0, %1\n\tv_nop" : "=v"(r) : "v"(x));
    return r;
#else
    return tanhf(x);
#endif
}

// Hardware v_cos_f32: input in revolutions (x/2pi). Valid here: |arg| <= pi,
// i.e. |rev| <= 0.5 -- no range reduction needed.
__device__ __forceinline__ float cos_fast(float x /*radians*/) {
#if defined(__AMDGCN__) && __has_builtin(__builtin_amdgcn_cosf)
    return __builtin_amdgcn_cosf(x * 0.15915494309189535f);
#else
    return cosf(x);
#endif
}

// 1-ulp hardware reciprocal / sqrt (one TRANS op instead of the IEEE-exact
// ~10-instruction div/sqrt fixup sequences); feeds acos/cos so 1 ulp is noise.
__device__ __forceinline__ float rcp_fast(float x) {
#if defined(__AMDGCN__) && __has_builtin(__builtin_amdgcn_rcpf)
    return __builtin_amdgcn_rcpf(x);
#else
    return 1.0f / x;
#endif
}
__device__ __forceinline__ float sqrt_fast(float x) {
#if defined(__AMDGCN__) && __has_builtin(__builtin_amdgcn_sqrtf)
    return __builtin_amdgcn_sqrtf(x);
#else
    return sqrtf(x);
#endif
}

// acos on [-1,1] via cephes asinf core: one sqrt + 5 FMA + selects.
// |err| ~ 1e-7 rad, far below the f32 eigenvalue noise of this pipeline.
__device__ __forceinline__ float acos_fast(float x) {
    const float ax  = fabsf(x);
    const bool  big = ax > 0.5f;
    const float z   = big ? 0.5f * (1.0f - ax) : ax * ax;
    const float s   = big ? sqrt_fast(z) : ax;
    float poly = fmaf(4.2163199048e-2f, z, 2.4181311049e-2f);
    poly = fmaf(poly, z, 4.5470025998e-2f);
    poly = fmaf(poly, z, 7.4953002686e-2f);
    poly = fmaf(poly, z, 1.6666752422e-1f);
    const float t = fmaf(s * z, poly, s);            // asin(s)
    const float r = big ? (2.0f * t) : (1.5707963267948966f - t); // acos(ax)
    return (x < 0.0f) ? (3.1415926535897931f - r) : r;
}

__global__ __launch_bounds__(256) void spectral_ode_kernel(
    const float* __restrict__ inp,   // B x 3 x 3 (exactly symmetric)
    const float* __restrict__ W1g,   // 6 x 1 x 3
    const float* __restrict__ W2g,   // 6 x 3 x 3
    const float* __restrict__ W3g,   // 6 x 3 x 1
    const float* __restrict__ bg,    // 6
    float* __restrict__ out,         // 6 x B
    int B)
{
    const int i = blockIdx.x * blockDim.x + threadIdx.x;
    if (i >= B) return;

    // ---- weights: wave-uniform -> scalar loads / SGPRs ----
    float w1[6][3], w2[6][3][3], w3dt[6][3], cb[6];
#pragma unroll
    for (int n = 0; n < 6; ++n) {
        cb[n] = DT_F * expf(bg[n]);
#pragma unroll
        for (int k = 0; k < 3; ++k) {
            w1[n][k]   = W1g[n * 3 + k];
            w3dt[n][k] = DT_F * W3g[n * 3 + k];
#pragma unroll
            for (int j = 0; j < 3; ++j)
                w2[n][k][j] = W2g[n * 9 + k * 3 + j];
        }
    }

    // ---- load symmetric 3x3 (upper triangle suffices) ----
    const float* p = inp + (size_t)i * 9;
    const float a00 = p[0], a01 = p[1], a02 = p[2];
    const float a11 = p[4], a12 = p[5], a22 = p[8];

    // ---- analytic eigenvalues (Smith's trigonometric method) ----
    const float q  = (a00 + a11 + a22) * (1.0f / 3.0f);
    const float p1 = a01 * a01 + a02 * a02 + a12 * a12;
    const float b00 = a00 - q, b11 = a11 - q, b22 = a22 - q;
    const float p2 = b00 * b00 + b11 * b11 + b22 * b22 + 2.0f * p1;
    const float pp = sqrt_fast(p2 * (1.0f / 6.0f));
    const float invp = rcp_fast(pp);

    const float c00 = b00 * invp, c01 = a01 * invp, c02 = a02 * invp;
    const float c11 = b11 * invp, c12 = a12 * invp, c22 = b22 * invp;
    float r = 0.5f * (c00 * (c11 * c22 - c12 * c12)
                    - c01 * (c01 * c22 - c12 * c02)
                    + c02 * (c01 * c12 - c11 * c02));
    r = fmaxf(-1.0f, fminf(1.0f, r));
    const float phi = acos_fast(r) * (1.0f / 3.0f);   // [0, pi/3]

    const float tau1 = q + 2.0f * pp * cos_fast(phi);                        // largest
    const float tau3 = q + 2.0f * pp * cos_fast(phi + 2.0943951023931953f);  // smallest
    const float tau2 = 3.0f * q - tau1 - tau3;

    // ---- initial states ----
    const float s   = tau1 + tau2 + tau3;
    const float n6v = tau1 * tau1 + tau2 * tau2 + tau3 * tau3
                    - tau1 * tau2 - tau1 * tau3 - tau2 * tau3;
    float st[6];
    st[0] = tau1;
    st[1] = tau1 + tau2;
    st[2] = s;
    st[3] = -tau3;
    st[4] = s * s;
    st[5] = n6v;

    // ---- 50 Euler steps of 6 independent 1->3->3->1 tanh MLPs ----
    // 6 independent channels give ILP for VOPD dual-issue and to hide
    // v_tanh TRANS latency. cb (= DT*exp(b)) folded into the W3 FMA chain.
    for (int t = 0; t < N_STEPS; ++t) {
#pragma unroll
        for (int n = 0; n < 6; ++n) {
            const float x  = st[n];
            const float h0 = tanh_fast(x * w1[n][0]);
            const float h1 = tanh_fast(x * w1[n][1]);
            const float h2 = tanh_fast(x * w1[n][2]);
            const float g0 = tanh_fast(fmaf(h0, w2[n][0][0], fmaf(h1, w2[n][1][0], h2 * w2[n][2][0])));
            const float g1 = tanh_fast(fmaf(h0, w2[n][0][1], fmaf(h1, w2[n][1][1], h2 * w2[n][2][1])));
            const float g2 = tanh_fast(fmaf(h0, w2[n][0][2], fmaf(h1, w2[n][1][2], h2 * w2[n][2][2])));
            st[n] = st[n] + fmaf(g0, w3dt[n][0],
                            fmaf(g1, w3dt[n][1],
                            fmaf(g2, w3dt[n][2], cb[n])));
        }
    }

    // ---- reconstruction:  M = sum d_i v_i v_i^T  ==  al*A^2 + be*A + ga*I
    // (Lagrange interpolation f(tau_i) = d_i; f64 once per sample to kill
    //  cancellation when eigenvalue gaps are small)
    const double T1 = tau1, T2 = tau2, T3 = tau3;
    const double S  = (double)s;
    const double N1 = st[0], N2 = st[1], N3 = st[2], N4 = st[3], N5 = st[4], N6 = st[5];
    const double twoN5s = 2.0 * N5 * S;
    const double d1 = N1 + N2 + N3 + twoN5s + N6 * (2.0 * T1 - T2 - T3);
    const double d2 =      N2 + N3 + twoN5s + N6 * (2.0 * T2 - T1 - T3);
    const double d3 =      N3 - N4 + twoN5s + N6 * (2.0 * T3 - T1 - T2);

    const double g12 = T1 - T2, g13 = T1 - T3, g23 = T2 - T3;
    const double D1 = g12 * g13;      // (T1-T2)(T1-T3)
    const double D2 = -g12 * g23;     // (T2-T1)(T2-T3)
    const double D3 = g13 * g23;      // (T3-T1)(T3-T2)
    const double inv = 1.0 / (D1 * D2 * D3);
    const double c1 = d1 * (D2 * D3) * inv;
    const double c2 = d2 * (D1 * D3) * inv;
    const double c3 = d3 * (D1 * D2) * inv;

    const double al = c1 + c2 + c3;
    const double be = -(c1 * (T2 + T3) + c2 * (T1 + T3) + c3 * (T1 + T2));
    const double ga = c1 * T2 * T3 + c2 * T1 * T3 + c3 * T1 * T2;

    const double A00 = a00, A01 = a01, A02 = a02, A11 = a11, A12 = a12, A22 = a22;
    const double S00 = A00 * A00 + A01 * A01 + A02 * A02;
    const double S01 = A00 * A01 + A01 * A11 + A02 * A12;
    const double S02 = A00 * A02 + A01 * A12 + A02 * A22;
    const double S11 = A01 * A01 + A11 * A11 + A12 * A12;
    const double S12 = A01 * A02 + A11 * A12 + A12 * A22;
    const double S22 = A02 * A02 + A12 * A12 + A22 * A22;

    const size_t ii = (size_t)i, Bs = (size_t)B;
    out[0 * Bs + ii] = (float)(al * S00 + be * A00 + ga);
    out[1 * Bs + ii] = (float)(al * S01 + be * A01);
    out[2 * Bs + ii] = (float)(al * S02 + be * A02);
    out[3 * Bs + ii] = (float)(al * S11 + be * A11 + ga);
    out[4 * Bs + ii] = (float)(al * S12 + be * A12);
    out[5 * Bs + ii] = (float)(al * S22 + be * A22 + ga);
}

extern "C" void kernel_launch(void* const* d_in, const int* in_sizes, int n_in,
                              void* d_out, int out_size, void* d_ws, size_t ws_size,
                              hipStream_t stream) {
    const float* inp = (const float*)d_in[0];
    const float* W1  = (const float*)d_in[1];
    const float* W2  = (const float*)d_in[2];
    const float* W3  = (const float*)d_in[3];
    const float* b   = (const float*)d_in[4];
    float* out       = (float*)d_out;

    const int B = in_sizes[0] / 9;
    const int threads = 256;                 // 8 wave32 waves per block
    const int blocks  = (B + threads - 1) / threads;
    hipLaunchKernelGGL(spectral_ode_kernel, dim3(blocks), dim3(threads), 0, stream,
                       inp, W1, W2, W3, b, out, B);
}